// ColorHead_87299505258635
// MI455X (gfx1250) — compile-verified
//
#include <hip/hip_runtime.h>
#include <hip/hip_bf16.h>

// ColorHead pipeline for MI455X (gfx1250, wave32).
//  k_zero          : zero stat accumulators (graph-replay safe)
//  k_mode_stage1   : 11x11 histogram-mode pool (17 bins, packed u8x4 counters,
//                    sliding-window: 121-update init + 22 updates/row)
//                    + fused 3->6 1x1 conv + leaky -> a[p][8] (K padded to 8)
//  k_stage2_wmma   : 6->32 GEMM via V_WMMA_F32_16X16X4_F32 (bias in C init),
//                    writes y[p][32], accumulates bn1 sum/sumsq
//  k_bn_final      : mean / rsqrt(var+eps)*gamma
//  k_t_stats       : sum/sumsq of t = leaky(bn1(y))*w3+b3  (bn2 stats)
//  k_down1         : fused bn1->leaky->dw->bn2->leaky + 3x3 stride-3 downsample
//  k_down          : generic depthwise 3x3 stride-3 pyramid levels
//  k_level_stats   : BN stats of nearest-upsampled z via multiplicity weights
//  k_level_final   : fold k_w/k_h + BN into per (level,channel) affine (A,B)
//  k_output        : score[b,co,i,j] = sum_l leaky(A*z_l + B), one 256MB write

typedef float v2f __attribute__((ext_vector_type(2)));
typedef float v8f __attribute__((ext_vector_type(8)));

#define HH 512
#define WW 512
#define HALF 32
#define NPIX (4*512*512)

#define MT_W 256            // tile width (one output column per thread)
#define MT_R 16             // output rows per thread (sliding window)
#define MT_CS 272           // LDS code row stride (bytes), width 266 padded

__global__ void k_zero(float* __restrict__ p, int n) {
    int i = blockIdx.x * blockDim.x + threadIdx.x;
    if (i < n) p[i] = 0.f;
}

// ---------------- mode pool (sliding window) + stage1 ----------------
__global__ void k_mode_stage1(const float* __restrict__ x,
                              const float* __restrict__ w1,
                              const float* __restrict__ b1,
                              float* __restrict__ a_out) {
    __shared__ unsigned char codes[(MT_R + 10) * MT_CS];
    __shared__ unsigned char modes[3 * MT_R * MT_W];   // self-read only
    const int tx = threadIdx.x;
    const int bx = blockIdx.x, by = blockIdx.y, b = blockIdx.z;
    const int x0 = bx * MT_W;
    const int y0 = by * MT_R;

    for (int c = 0; c < 3; ++c) {
        __syncthreads();   // codes[] reused across channels
        // load (R+10) x (W+10) quantized codes with reflect-pad 5
        for (int idx = tx; idx < (MT_R + 10) * (MT_W + 10); idx += 256) {
            int ly = idx / (MT_W + 10);
            int lx = idx - ly * (MT_W + 10);
            int gy = y0 + ly - 5;
            if (gy < 0) gy = -gy;
            if (gy >= HH) gy = 2 * HH - 2 - gy;
            int gx = x0 + lx - 5;
            if (gx < 0) gx = -gx;
            if (gx >= WW) gx = 2 * WW - 2 - gx;
            float v = x[(((long)b * 3 + c) * HH + gy) * WW + gx];
            float q = rintf(v * 15.9375f);          // round(x*255/16), half-even
            q = fminf(fmaxf(q, 0.f), 16.f);
            codes[ly * MT_CS + lx] = (unsigned char)(int)q;
        }
        __syncthreads();

        // 17-bin histogram packed into 5 u32s (8-bit fields, max count 121)
        unsigned int h0 = 0, h1 = 0, h2 = 0, h3 = 0, h4 = 0;
        for (int u = 0; u < 11; ++u) {              // init: window rows 0..10
            #pragma unroll
            for (int v = 0; v < 11; ++v) {
                int code = codes[u * MT_CS + tx + v];
                unsigned int inc = 1u << ((code & 3) * 8);
                int wsel = code >> 2;
                h0 += (wsel == 0) ? inc : 0u;
                h1 += (wsel == 1) ? inc : 0u;
                h2 += (wsel == 2) ? inc : 0u;
                h3 += (wsel == 3) ? inc : 0u;
                h4 += (wsel == 4) ? inc : 0u;
            }
        }
        for (int r = 0;; ++r) {
            // argmax, ties -> smallest bin: key = (count<<5)|(16-bin)
            unsigned int hw[5] = {h0, h1, h2, h3, h4};
            unsigned int best = 0;
            #pragma unroll
            for (int bin = 0; bin < 17; ++bin) {
                unsigned int cnt = (hw[bin >> 2] >> ((bin & 3) * 8)) & 255u;
                unsigned int key = (cnt << 5) | (unsigned)(16 - bin);
                best = key > best ? key : best;
            }
            modes[(c * MT_R + r) * MT_W + tx] = (unsigned char)(16 - (int)(best & 31u));
            if (r == MT_R - 1) break;
            // slide down one row: remove LDS row r, add LDS row r+11.
            // Counts are exact so 8-bit fields never borrow/carry across.
            #pragma unroll
            for (int v = 0; v < 11; ++v) {
                int cd = codes[r * MT_CS + tx + v];
                unsigned int dec = 1u << ((cd & 3) * 8);
                int wd = cd >> 2;
                h0 -= (wd == 0) ? dec : 0u;
                h1 -= (wd == 1) ? dec : 0u;
                h2 -= (wd == 2) ? dec : 0u;
                h3 -= (wd == 3) ? dec : 0u;
                h4 -= (wd == 4) ? dec : 0u;
                int ca = codes[(r + 11) * MT_CS + tx + v];
                unsigned int ai = 1u << ((ca & 3) * 8);
                int wa = ca >> 2;
                h0 += (wa == 0) ? ai : 0u;
                h1 += (wa == 1) ? ai : 0u;
                h2 += (wa == 2) ? ai : 0u;
                h3 += (wa == 3) ? ai : 0u;
                h4 += (wa == 4) ? ai : 0u;
            }
        }
    }
    __syncthreads();
    // stage1: 3->6 1x1 conv + leaky; a[p][8], K-pad zeros at k=6,7
    float W1[6][3], B1[6];
    #pragma unroll
    for (int o = 0; o < 6; ++o) {
        W1[o][0] = w1[o * 3];
        W1[o][1] = w1[o * 3 + 1];
        W1[o][2] = w1[o * 3 + 2];
        B1[o] = b1[o];
    }
    for (int r = 0; r < MT_R; ++r) {
        float m0 = (float)modes[(0 * MT_R + r) * MT_W + tx] * 0.0625f;
        float m1 = (float)modes[(1 * MT_R + r) * MT_W + tx] * 0.0625f;
        float m2 = (float)modes[(2 * MT_R + r) * MT_W + tx] * 0.0625f;
        long p = ((long)b * HH + (y0 + r)) * WW + (x0 + tx);
        float* ap = a_out + p * 8;
        #pragma unroll
        for (int o = 0; o < 6; ++o) {
            float s = B1[o] + m0 * W1[o][0] + m1 * W1[o][1] + m2 * W1[o][2];
            ap[o] = s > 0.f ? s : 0.01f * s;
        }
        ap[6] = 0.f;
        ap[7] = 0.f;
    }
}

// ---------------- stage2: 6->32 GEMM via WMMA f32 16x16x4 ----------------
__global__ void k_stage2_wmma(const float* __restrict__ a_in,
                              const float* __restrict__ w2,
                              const float* __restrict__ b2,
                              float* __restrict__ y_out,
                              float* __restrict__ stats) {
    const int NGROUP = NPIX / 16;                      // 65536 groups of 16 pixels
    int lane = threadIdx.x & 31;
    int wid = (blockIdx.x * blockDim.x + threadIdx.x) >> 5;
    int nwaves = (gridDim.x * blockDim.x) >> 5;
    int m = lane & 15;
    int khalf = lane >> 4;                             // 0: K{0,1}, 1: K{2,3}
    // B operands (K x N = 4x16 per op): lane-half selects K pair, VGPR0/1 = K even/odd
    int kb = khalf * 2;
    float B00x = w2[m * 6 + kb],     B00y = w2[m * 6 + kb + 1];
    float B01x = (kb + 4 < 6) ? w2[m * 6 + kb + 4] : 0.f;
    float B01y = (kb + 5 < 6) ? w2[m * 6 + kb + 5] : 0.f;
    float B10x = w2[(m + 16) * 6 + kb],     B10y = w2[(m + 16) * 6 + kb + 1];
    float B11x = (kb + 4 < 6) ? w2[(m + 16) * 6 + kb + 4] : 0.f;
    float B11y = (kb + 5 < 6) ? w2[(m + 16) * 6 + kb + 5] : 0.f;
    v2f B00 = {B00x, B00y}, B01 = {B01x, B01y}, B10 = {B10x, B10y}, B11 = {B11x, B11y};
    float bias0 = b2[m], bias1 = b2[m + 16];
    float s0 = 0.f, ss0 = 0.f, s1 = 0.f, ss1 = 0.f;
    for (int g = wid; g < NGROUP; g += nwaves) {
        const float* ap = a_in + (long)g * 16 * 8 + m * 8 + khalf * 2;
        v2f A0 = {ap[0], ap[1]};                       // K slice 0 (k = 0..3)
        v2f A1 = {ap[4], ap[5]};                       // K slice 1 (k = 4..7, padded)
        int gn = g + nwaves;                           // wave-uniform guard: EXEC stays all-1s
        if (gn < NGROUP)                               // prefetch next strided A group
            __builtin_prefetch((const char*)a_in + (long)gn * 512 + lane * 16, 0, 1);
        v8f c0 = {bias0, bias0, bias0, bias0, bias0, bias0, bias0, bias0};
        v8f c1 = {bias1, bias1, bias1, bias1, bias1, bias1, bias1, bias1};
        c0 = __builtin_amdgcn_wmma_f32_16x16x4_f32(false, A0, false, B00, (short)0, c0, false, false);
        c0 = __builtin_amdgcn_wmma_f32_16x16x4_f32(false, A1, false, B01, (short)0, c0, false, false);
        c1 = __builtin_amdgcn_wmma_f32_16x16x4_f32(false, A0, false, B10, (short)0, c1, false, false);
        c1 = __builtin_amdgcn_wmma_f32_16x16x4_f32(false, A1, false, B11, (short)0, c1, false, false);
        float* yb = y_out + (long)g * 16 * 32;
        #pragma unroll
        for (int r = 0; r < 8; ++r) {                  // C: VGPR r, lane-half -> M=r / r+8
            int mm = r + khalf * 8;
            float v0 = c0[r], v1 = c1[r];
            yb[mm * 32 + m] = v0;
            yb[mm * 32 + m + 16] = v1;
            s0 += v0; ss0 += v0 * v0;
            s1 += v1; ss1 += v1 * v1;
        }
    }
    atomicAdd(&stats[m], s0);
    atomicAdd(&stats[32 + m], ss0);
    atomicAdd(&stats[m + 16], s1);
    atomicAdd(&stats[32 + m + 16], ss1);
}

// ---------------- BN finalize (mean, rsqrt(var+eps)*gamma) ----------------
__global__ void k_bn_final(const float* __restrict__ sums, const float* __restrict__ g,
                           float* __restrict__ m_out, float* __restrict__ A_out, float N) {
    int c = threadIdx.x;
    float m = sums[c] / N;
    float var = sums[32 + c] / N - m * m;
    var = var > 0.f ? var : 0.f;
    m_out[c] = m;
    A_out[c] = rsqrtf(var + 1e-5f) * g[c];
}

// ---------------- bn2 stats over t = leaky(bn1(y))*w3 + b3 ----------------
__global__ void k_t_stats(const float* __restrict__ y, const float* __restrict__ stats_in,
                          const float* __restrict__ bn1b, const float* __restrict__ w3,
                          const float* __restrict__ b3, float* __restrict__ stats_out) {
    __shared__ float ssum[32], sss[32];
    if (threadIdx.x < 32) { ssum[threadIdx.x] = 0.f; sss[threadIdx.x] = 0.f; }
    __syncthreads();
    const long NE = (long)NPIX * 32;
    int c = threadIdx.x & 31;                          // fixed: stride multiple of 32
    float m1 = stats_in[128 + c], A1 = stats_in[160 + c], bb = bn1b[c];
    float wc = w3[c], bc = b3[c];
    long stride = (long)gridDim.x * blockDim.x;
    float s = 0.f, ss = 0.f;
    for (long e = (long)blockIdx.x * blockDim.x + threadIdx.x; e < NE; e += stride) {
        float u = (y[e] - m1) * A1 + bb;
        u = u > 0.f ? u : 0.01f * u;
        float t = u * wc + bc;
        s += t; ss += t * t;
    }
    atomicAdd(&ssum[c], s);
    atomicAdd(&sss[c], ss);
    __syncthreads();
    if (threadIdx.x < 32) {
        atomicAdd(&stats_out[64 + threadIdx.x], ssum[threadIdx.x]);
        atomicAdd(&stats_out[96 + threadIdx.x], sss[threadIdx.x]);
    }
}

// ---- fused bn1->leaky->dw->bn2->leaky + level-1 downsample (512 -> 171) ----
__global__ void k_down1(const float* __restrict__ y, const float* __restrict__ stats,
                        const float* __restrict__ bn1b, const float* __restrict__ w3,
                        const float* __restrict__ b3, const float* __restrict__ bn2b,
                        const float* __restrict__ kd, float* __restrict__ d1) {
    const int HO = 171;
    int lane = threadIdx.x & 31;                       // lane = channel
    int slot = (blockIdx.x * blockDim.x + threadIdx.x) >> 5;
    if (slot >= 4 * HO * HO) return;
    int j = slot % HO, i = (slot / HO) % HO, b = slot / (HO * HO);
    float m1 = stats[128 + lane], A1 = stats[160 + lane], bb1 = bn1b[lane];
    float m2 = stats[192 + lane], A2 = stats[224 + lane], bb2 = bn2b[lane];
    float wc = w3[lane], bc = b3[lane];
    float acc = 0.f;
    #pragma unroll
    for (int u = 0; u < 3; ++u) {
        int r = 3 * i - 1 + u;
        if (r < 0 || r >= HH) continue;
        #pragma unroll
        for (int v = 0; v < 3; ++v) {
            int cc = 3 * j - 1 + v;
            if (cc < 0 || cc >= WW) continue;
            float yv = y[(((long)b * HH + r) * WW + cc) * 32 + lane];
            float uu = (yv - m1) * A1 + bb1;
            uu = uu > 0.f ? uu : 0.01f * uu;
            float t = uu * wc + bc;
            float hv = (t - m2) * A2 + bb2;
            hv = hv > 0.f ? hv : 0.01f * hv;
            acc += hv * kd[lane * 9 + u * 3 + v];
        }
    }
    d1[(((long)b * 32 + lane) * HO + i) * HO + j] = acc;
}

// ------------- generic depthwise 3x3 stride-3 pad-1 downsample -------------
__global__ void k_down(const float* __restrict__ din, float* __restrict__ dout,
                       const float* __restrict__ kd, int hi, int ho) {
    int total = 4 * 32 * ho * ho;
    int gid = blockIdx.x * blockDim.x + threadIdx.x;
    if (gid >= total) return;
    int j = gid % ho, i = (gid / ho) % ho, bc = gid / (ho * ho), c = bc & 31;
    float acc = 0.f;
    #pragma unroll
    for (int u = 0; u < 3; ++u) {
        int r = 3 * i - 1 + u;
        if (r < 0 || r >= hi) continue;
        #pragma unroll
        for (int v = 0; v < 3; ++v) {
            int cc = 3 * j - 1 + v;
            if (cc < 0 || cc >= hi) continue;
            acc += din[((long)bc * hi + r) * hi + cc] * kd[c * 9 + u * 3 + v];
        }
    }
    dout[((long)bc * ho + i) * ho + j] = acc;
}

// -------- BN stats of nearest-upsampled z via multiplicity weights --------
__global__ void k_level_stats(const float* __restrict__ d, float* __restrict__ stats,
                              int h, int lvl) {
    __shared__ float ssum[32], sss[32];
    if (threadIdx.x < 32) { ssum[threadIdx.x] = 0.f; sss[threadIdx.x] = 0.f; }
    __syncthreads();
    int total = 4 * 32 * h * h;
    for (int gid = blockIdx.x * blockDim.x + threadIdx.x; gid < total;
         gid += gridDim.x * blockDim.x) {
        int j = gid % h, i = (gid / h) % h, c = (gid / (h * h)) & 31;
        // multiplicity of src row r under dst->src map (i*h)//512
        float mr = (float)((512 * (i + 1) + h - 1) / h - (512 * i + h - 1) / h);
        float mc = (float)((512 * (j + 1) + h - 1) / h - (512 * j + h - 1) / h);
        float wv = mr * mc;
        float v = d[gid];
        atomicAdd(&ssum[c], wv * v);
        atomicAdd(&sss[c], wv * v * v);
    }
    __syncthreads();
    if (threadIdx.x < 32) {
        atomicAdd(&stats[256 + lvl * 32 + threadIdx.x], ssum[threadIdx.x]);
        atomicAdd(&stats[448 + lvl * 32 + threadIdx.x], sss[threadIdx.x]);
    }
}

// ---- fold k_w/k_h and branch BN into affine: branch = leaky(A*z + B) ----
__global__ void k_level_final(float* __restrict__ stats,
                              const float* __restrict__ kw, const float* __restrict__ gw,
                              const float* __restrict__ bw,
                              const float* __restrict__ kh, const float* __restrict__ gh,
                              const float* __restrict__ bh) {
    int t = threadIdx.x;
    if (t >= 192) return;
    int c = t & 31;
    const float N = 4.f * 512.f * 512.f;
    float S = stats[256 + t], SS = stats[448 + t];
    float mean = S / N;
    float var = SS / N - mean * mean;
    var = var > 0.f ? var : 0.f;
    float kwc = kw[c];
    float sw = rsqrtf(kwc * kwc * var + 1e-5f);        // var(z*k) = k^2 var(z)
    stats[640 + t] = kwc * sw * gw[c];
    stats[832 + t] = -kwc * mean * sw * gw[c] + bw[c];
    float khc = kh[c];
    float sh = rsqrtf(khc * khc * var + 1e-5f);
    stats[1024 + t] = khc * sh * gh[c];
    stats[1216 + t] = -khc * mean * sh * gh[c] + bh[c];
}

// ---------------- final fused output pass (single 256MB write) ----------------
__global__ void k_output(const float* __restrict__ dpyr, const float* __restrict__ stats,
                         float* __restrict__ out) {
    const int HS[6] = {171, 57, 19, 7, 3, 1};
    const long OFF[6] = {0, 3742848, 4158720, 4204928, 4211200, 4212352};
    long gid = (long)blockIdx.x * blockDim.x + threadIdx.x;
    int j = (int)(gid & 511);
    int i = (int)((gid >> 9) & 511);
    int co = (int)((gid >> 18) & 63);
    int b = (int)(gid >> 24);
    int c = co & 31, br = co >> 5;                     // 0..31 width branch, 32..63 height
    const float* Ap = stats + (br ? 1024 : 640);
    const float* Bp = stats + (br ? 1216 : 832);
    float s = 0.f;
    #pragma unroll
    for (int l = 0; l < 6; ++l) {
        int h = HS[l];
        int ri = (i * h) >> 9, cj = (j * h) >> 9;      // nearest: (dst*in)//512
        float z = dpyr[OFF[l] + (((long)b * 32 + c) * h + ri) * h + cj];
        float v = Ap[l * 32 + c] * z + Bp[l * 32 + c];
        s += v > 0.f ? v : 0.01f * v;
    }
    out[gid] = s;
}

extern "C" void kernel_launch(void* const* d_in, const int* in_sizes, int n_in,
                              void* d_out, int out_size, void* d_ws, size_t ws_size,
                              hipStream_t stream) {
    const float* x    = (const float*)d_in[0];
    const float* w1   = (const float*)d_in[1];
    const float* b1   = (const float*)d_in[2];
    const float* w2   = (const float*)d_in[3];
    const float* b2   = (const float*)d_in[4];
    const float* bn1g = (const float*)d_in[5];
    const float* bn1b = (const float*)d_in[6];
    const float* w3   = (const float*)d_in[7];
    const float* b3   = (const float*)d_in[8];
    const float* bn2g = (const float*)d_in[9];
    const float* bn2b = (const float*)d_in[10];
    const float* kdwn = (const float*)d_in[11];
    const float* kw   = (const float*)d_in[12];
    const float* bnwg = (const float*)d_in[13];
    const float* bnwb = (const float*)d_in[14];
    const float* kh   = (const float*)d_in[15];
    const float* bnhg = (const float*)d_in[16];
    const float* bnhb = (const float*)d_in[17];

    // workspace layout (bytes):
    //   a      [P*8  f32]  @ 0           (33,554,432)
    //   y      [P*32 f32]  @ 33,554,432  (134,217,728)
    //   d pyr  [4,212,480 f32] @ 167,772,160 (16,849,920)
    //   stats  [1408 f32]  @ 184,622,080
    char* ws = (char*)d_ws;
    float* a_ws  = (float*)(ws);
    float* y_ws  = (float*)(ws + 33554432);
    float* d_pyr = (float*)(ws + 167772160);
    float* stats = (float*)(ws + 184622080);

    const float NPIXF = (float)NPIX;
    const int HS[6] = {171, 57, 19, 7, 3, 1};
    const long OFF[6] = {0, 3742848, 4158720, 4204928, 4211200, 4212352};

    k_zero<<<4, 256, 0, stream>>>(stats, 640);
    k_mode_stage1<<<dim3(WW / MT_W, HH / MT_R, 4), 256, 0, stream>>>(x, w1, b1, a_ws);
    k_stage2_wmma<<<512, 256, 0, stream>>>(a_ws, w2, b2, y_ws, stats);
    k_bn_final<<<1, 32, 0, stream>>>(stats + 0, bn1g, stats + 128, stats + 160, NPIXF);
    k_t_stats<<<2048, 256, 0, stream>>>(y_ws, stats, bn1b, w3, b3, stats);
    k_bn_final<<<1, 32, 0, stream>>>(stats + 64, bn2g, stats + 192, stats + 224, NPIXF);
    {
        int slots = 4 * 171 * 171;
        int blocks = (slots * 32 + 255) / 256;
        k_down1<<<blocks, 256, 0, stream>>>(y_ws, stats, bn1b, w3, b3, bn2b, kdwn, d_pyr);
    }
    for (int l = 1; l < 6; ++l) {
        int total = 4 * 32 * HS[l] * HS[l];
        k_down<<<(total + 255) / 256, 256, 0, stream>>>(d_pyr + OFF[l - 1], d_pyr + OFF[l],
                                                        kdwn, HS[l - 1], HS[l]);
    }
    for (int l = 0; l < 6; ++l) {
        int total = 4 * 32 * HS[l] * HS[l];
        k_level_stats<<<(total + 255) / 256, 256, 0, stream>>>(d_pyr + OFF[l], stats, HS[l], l);
    }
    k_level_final<<<1, 192, 0, stream>>>(stats, kw, bnwg, bnwb, kh, bnhg, bnhb);
    k_output<<<262144, 256, 0, stream>>>(d_pyr, stats, (float*)d_out);
}